// PeripheralEnvisionate_74818330296856
// MI455X (gfx1250) — compile-verified
//
#include <hip/hip_runtime.h>
#include <hip/hip_bf16.h>
#include <math.h>

// Geometry (fixed by reference constants for x = (32,3,1080,1920)):
//   BC = 32*3 = 96 images of 1080x1920 -> 224x224
//   Horizontal bands (out col -> src region), all resize 1080 rows -> 224 rows vertically.
#define BC      96
#define H_IN    1080
#define W_IN    1920
#define OUT     224
#define HTAPS   48            // max horizontal taps (318->15 needs <=43)
#define VK      96            // padded K window per 16-output-row tile (span <= ~83)
#define NTILES_R (OUT/16)     // 14
#define NTILES_C (OUT/16)     // 14

typedef __attribute__((ext_vector_type(2))) float v2f;
typedef __attribute__((ext_vector_type(8))) float v8f;

// jax.image.resize(method='linear', antialias=True) weights:
//   inv = in/out, ks = max(inv,1), sf = (i+0.5)*inv - 0.5
//   w(j) = max(0, 1 - |j - sf|/ks), normalized over in-range taps.
__device__ inline void resize_weights(int i, int in_w, int out_w, float* w,
                                      int* base, int* cnt, int maxtaps) {
    float inv = (float)in_w / (float)out_w;
    float ks  = inv > 1.0f ? inv : 1.0f;
    float sf  = ((float)i + 0.5f) * inv - 0.5f;
    int lo = (int)ceilf(sf - ks);  if (lo < 0) lo = 0;
    int hi = (int)floorf(sf + ks); if (hi > in_w - 1) hi = in_w - 1;
    int n = hi - lo + 1; if (n > maxtaps) n = maxtaps; if (n < 1) n = 1;
    float s = 0.0f;
    for (int t = 0; t < n; ++t) {
        float d  = fabsf((float)(lo + t) - sf) / ks;
        float wt = 1.0f - d; if (wt < 0.0f) wt = 0.0f;
        w[t] = wt; s += wt;
    }
    float r = (s != 0.0f) ? 1.0f / s : 0.0f;
    for (int t = 0; t < n; ++t) w[t] *= r;
    *base = lo; *cnt = n;
}

// Kernel 0: build horizontal per-column tables and banded vertical weight tiles.
__global__ __launch_bounds__(256)
void pe_weights_kernel(float* __restrict__ hW, int* __restrict__ hBase,
                       int* __restrict__ hCnt, float* __restrict__ vW,
                       int* __restrict__ vBase) {
    int tid = threadIdx.x;
    for (int i = tid; i < NTILES_R * 16 * VK; i += 256) vW[i] = 0.0f;
    __syncthreads();
    if (tid < 14) { // per-tile vertical K-window base
        float inv = (float)H_IN / (float)OUT;
        float sf0 = ((float)(tid * 16) + 0.5f) * inv - 0.5f;
        int kb = (int)ceilf(sf0 - inv);
        if (kb < 0) kb = 0;
        if (kb > H_IN - VK) kb = H_IN - VK;
        vBase[tid] = kb;
    }
    if (tid < OUT) {
        // ---- horizontal band table for out column tid ----
        int ox = tid, in_w, out_w, srcoff, oxl;
        if      (ox < 15)  { in_w = 318;  out_w = 15;  srcoff = 0;    oxl = ox;       }
        else if (ox < 44)  { in_w = 318;  out_w = 29;  srcoff = 318;  oxl = ox - 15;  }
        else if (ox < 178) { in_w = 1080; out_w = 224; srcoff = 420;  oxl = ox;       }
        else if (ox < 209) { in_w = 318;  out_w = 31;  srcoff = 1284; oxl = ox - 178; }
        else               { in_w = 318;  out_w = 15;  srcoff = 1602; oxl = ox - 209; }
        float w[HTAPS]; int base, cnt;
        resize_weights(oxl, in_w, out_w, w, &base, &cnt, HTAPS);
        hBase[ox] = srcoff + base;
        hCnt[ox]  = cnt;
        for (int t = 0; t < HTAPS; ++t) hW[ox * HTAPS + t] = (t < cnt) ? w[t] : 0.0f;
        // ---- vertical weights scattered into banded tile (16 x VK) ----
        int oy = tid;
        float vw[16]; int vb, vc;
        resize_weights(oy, H_IN, OUT, vw, &vb, &vc, 16);
        int tr = oy >> 4;
        float inv = (float)H_IN / (float)OUT;
        float sf0 = ((float)(tr * 16) + 0.5f) * inv - 0.5f;
        int kb = (int)ceilf(sf0 - inv);
        if (kb < 0) kb = 0;
        if (kb > H_IN - VK) kb = H_IN - VK;
        for (int k = 0; k < vc; ++k) {
            int idx = vb + k - kb;
            if (idx >= 0 && idx < VK)
                vW[(tr * 16 + (oy & 15)) * VK + idx] = vw[k];
        }
    }
}

// Kernel 1: horizontal pass. One block per (bc,row); stage the full 1920-col
// row in LDS with b128 loads, then 224 threads each reduce one output column.
// x (96,1080,1920) -> tmpH (96,1080,224)
__global__ __launch_bounds__(256)
void pe_hpass_kernel(const float* __restrict__ x, float* __restrict__ tmpH,
                     const float* __restrict__ hW, const int* __restrict__ hBase,
                     const int* __restrict__ hCnt) {
    __shared__ __align__(16) float row[W_IN];
    int rid = blockIdx.x;                         // 0 .. BC*H_IN-1
    const float4* src4 = (const float4*)(x + (size_t)rid * W_IN);
    float4* row4 = (float4*)row;
    for (int i = threadIdx.x; i < W_IN / 4; i += 256) row4[i] = src4[i];
    __syncthreads();
    int ox = threadIdx.x;
    if (ox < OUT) {
        int b = hBase[ox], n = hCnt[ox];
        const float* wp = hW + ox * HTAPS;
        float acc = 0.0f;
        for (int t = 0; t < n; ++t) acc = fmaf(wp[t], row[b + t], acc);
        tmpH[(size_t)rid * OUT + ox] = acc;
    }
}

// Kernel 2: vertical pass as banded fp32 WMMA GEMM.
// out16x16(bc,tr,tc) = vWtile(16 x 96) @ tmpH[bc][kb..kb+95][ox0..ox0+15]
// using v_wmma_f32_16x16x4_f32, K stepped by 4. All lanes active (EXEC req).
__global__ __launch_bounds__(256)
void pe_vpass_wmma_kernel(const float* __restrict__ tmpH, float* __restrict__ out,
                          const float* __restrict__ vW, const int* __restrict__ vBase) {
    int wave = threadIdx.x >> 5;
    int lane = threadIdx.x & 31;
    int tile = blockIdx.x * 8 + wave;             // 0 .. 96*14*14-1 (exact fit)
    int bc   = tile / (NTILES_R * NTILES_C);
    int rem  = tile - bc * (NTILES_R * NTILES_C);
    int tr   = rem / NTILES_C;
    int tc   = rem - tr * NTILES_C;
    int oy0  = tr * 16, ox0 = tc * 16;
    int kb   = vBase[tr];
    int half = lane >> 4;                          // 0: K+0/K+1, 1: K+2/K+3
    int l15  = lane & 15;

    // A 16x4 f32 layout: lanes0-15 M=l15 {K=k,k+1}; lanes16-31 M=l15 {K=k+2,k+3}
    const float* arow = vW + (tr * 16 + l15) * VK + half * 2;
    // B 4x16 f32 layout: VGPR0 lanes0-15 K=k (16-31: K=k+2); VGPR1 K=k+1 (K=k+3)
    const float* bcol = tmpH + ((size_t)bc * H_IN + kb + half * 2) * OUT + ox0 + l15;

    v8f c = {0.f, 0.f, 0.f, 0.f, 0.f, 0.f, 0.f, 0.f};
#pragma unroll 4
    for (int k = 0; k < VK; k += 4) {
        v2f a, b;
        a.x = arow[k];
        a.y = arow[k + 1];
        b.x = bcol[0];
        b.y = bcol[OUT];
        bcol += 4 * OUT;
        c = __builtin_amdgcn_wmma_f32_16x16x4_f32(
                /*neg_a=*/false, a, /*neg_b=*/false, b,
                /*c_mod=*/(short)0, c, /*reuse_a=*/false, /*reuse_b=*/false);
    }
    // C/D 16x16 f32: VGPR r -> M=r (lanes0-15) / M=r+8 (lanes16-31), N=l15
    float* op = out + ((size_t)bc * OUT + oy0 + half * 8) * OUT + ox0 + l15;
#pragma unroll
    for (int r = 0; r < 8; ++r) op[r * OUT] = c[r];
}

extern "C" void kernel_launch(void* const* d_in, const int* in_sizes, int n_in,
                              void* d_out, int out_size, void* d_ws, size_t ws_size,
                              hipStream_t stream) {
    (void)in_sizes; (void)n_in; (void)out_size; (void)ws_size;
    const float* x = (const float*)d_in[0];
    float* out = (float*)d_out;

    char* ws = (char*)d_ws;
    size_t off = 0;
    float* tmpH  = (float*)(ws + off); off += (size_t)BC * H_IN * OUT * sizeof(float); // 92.9 MB
    float* hW    = (float*)(ws + off); off += (size_t)OUT * HTAPS * sizeof(float);
    int*   hBase = (int*)  (ws + off); off += (size_t)OUT * sizeof(int);
    int*   hCnt  = (int*)  (ws + off); off += (size_t)OUT * sizeof(int);
    float* vW    = (float*)(ws + off); off += (size_t)NTILES_R * 16 * VK * sizeof(float);
    int*   vBase = (int*)  (ws + off); off += (size_t)NTILES_R * sizeof(int);

    pe_weights_kernel<<<1, 256, 0, stream>>>(hW, hBase, hCnt, vW, vBase);
    pe_hpass_kernel<<<BC * H_IN, 256, 0, stream>>>(x, tmpH, hW, hBase, hCnt);
    pe_vpass_wmma_kernel<<<(BC * NTILES_R * NTILES_C) / 8, 256, 0, stream>>>(tmpH, out, vW, vBase);
}